// Conv1dfftAutograd_53953379172722
// MI455X (gfx1250) — compile-verified
//
#include <hip/hip_runtime.h>

typedef __attribute__((ext_vector_type(2))) float v2f;
typedef __attribute__((ext_vector_type(8))) float v8f;

#define N_BATCH 64
#define C_IN    64
#define W_IN    4096
#define F_OUT   64
#define KW      16
#define OUT_W   (W_IN - KW + 1)      // 4081
#define TILE_T  256
#define CHUNK_C 4
#define CHUNK_K (CHUNK_C * KW)       // 64 K-columns staged per iteration
#define N_CHUNK (C_IN / CHUNK_C)     // 16
#define XS_PITCH 272                 // 256 + 15 halo, padded to x16B

#if __has_builtin(__builtin_amdgcn_s_wait_asynccnt)
#define WAIT_ASYNC(n) __builtin_amdgcn_s_wait_asynccnt(n)
#else
#define WAIT_ASYNC(n) asm volatile("s_wait_asynccnt %0" :: "i"(n) : "memory")
#endif

// CDNA5 async copy: global -> LDS, 16B per active lane, tracked by ASYNCcnt.
__device__ __forceinline__ void async_b128(unsigned lds_off, const void* gaddr) {
    asm volatile("global_load_async_to_lds_b128 %0, %1, off"
                 :: "v"(lds_off), "v"((unsigned long long)(uintptr_t)gaddr)
                 : "memory");
}

// out[n,f,t] = sum_{c,k} x[n,c,t+k] * w[f,c,k] + b[f]
// Implicit GEMM per n: O(64 x 4081) = W(64 x 1024) * im2col(X)(1024 x 4081),
// V_WMMA_F32_16X16X4_F32 accumulation (full fp32), double-buffered
// GLOBAL_LOAD_ASYNC_TO_LDS_B128 staging overlapped with compute.
__global__ __launch_bounds__(256) void conv1d_wmma_f32_kernel(
    const float* __restrict__ x, const float* __restrict__ w,
    const float* __restrict__ b, float* __restrict__ out)
{
    __shared__ alignas(16) float ws[2][F_OUT * CHUNK_K];     // 2 x 16 KB
    __shared__ alignas(16) float xs[2][CHUNK_C * XS_PITCH];  // 2 x 4.25 KB

    const int t0   = blockIdx.x * TILE_T;   // 16 tiles cover t in [0,4096)
    const int n    = blockIdx.y;
    const int tid  = threadIdx.x;
    const int lane = tid & 31;              // wave32
    const int wave = tid >> 5;              // 8 waves
    const int half = lane >> 4;
    const int lrow = lane & 15;

    const int fr = (wave & 3) * 16;         // this wave's 16-row F strip
    const int tb = (wave >> 2) * 128;       // this wave's 128-col T strip

    const float* xrow = x + (size_t)n * C_IN * W_IN;

    // LDS byte offsets (generic LDS pointer low 32 bits == LDS offset)
    const unsigned wsb[2] = { (unsigned)(uintptr_t)&ws[0][0],
                              (unsigned)(uintptr_t)&ws[1][0] };
    const unsigned xsb[2] = { (unsigned)(uintptr_t)&xs[0][0],
                              (unsigned)(uintptr_t)&xs[1][0] };

    // Issue one chunk's staging: exactly 6 async ops per wave (uniform, so
    // s_wait_asynccnt 6 later means "previous batch fully landed").
    auto issue = [&](int chunk, int buf) {
        const int c0 = chunk * CHUNK_C;
        // w tile: 1024 float4, 4 per lane. ws[f][ck] = w[f, chunk, ck]
        #pragma unroll
        for (int i = 0; i < 4; ++i) {
            const int idx = wave * 128 + i * 32 + lane;   // 0..1023
            const int f = idx >> 4;
            const int q = (idx & 15) << 2;
            async_b128(wsb[buf] + (unsigned)idx * 16,
                       w + (size_t)f * (C_IN * KW) + chunk * CHUNK_K + q);
        }
        // x tile: 272 float4, 34 per wave (2nd issue pads lanes with
        // duplicate addresses so all 32 lanes stay active).
        #pragma unroll
        for (int i = 0; i < 2; ++i) {
            const int j = wave * 34 + (i == 0 ? lane : 32 + (lane & 1));
            const int cl = j / 68;
            const int q4 = (j - cl * 68) << 2;
            int gq = t0 + q4;
            if (gq > W_IN - 4) gq = W_IN - 4;   // clamp: tail feeds only t>4080
            async_b128(xsb[buf] + (unsigned)j * 16,
                       xrow + (size_t)(c0 + cl) * W_IN + gq);
        }
    };

    v8f acc[8] = {};                        // 8 tiles of 16x16 f32 accumulators

    issue(0, 0);
    for (int chunk = 0; chunk < N_CHUNK; ++chunk) {
        const int cur = chunk & 1;
        if (chunk + 1 < N_CHUNK) {
            issue(chunk + 1, cur ^ 1);      // prefetch next chunk (other buffer)
            WAIT_ASYNC(6);                  // current chunk's 6 ops have landed
        } else {
            WAIT_ASYNC(0);
        }
        __syncthreads();                    // all waves' tiles visible

        const float* wsc = ws[cur];
        const float* xsc = xs[cur];
        for (int cl = 0; cl < CHUNK_C; ++cl) {
            const float* xc = xsc + cl * XS_PITCH;
            #pragma unroll
            for (int kk = 0; kk < KW; kk += 4) {
                // A (16x4 f32, 2 VGPRs): lane -> A[fr+lrow][kk + 2*half +{0,1}]
                v2f a = *(const v2f*)(wsc + (fr + lrow) * CHUNK_K
                                          + cl * KW + kk + 2 * half);
                // Batch ALL 8 B-operand loads first so the DS round-trip
                // latency overlaps the XDL pipe (staggered s_wait_dscnt),
                // instead of a load->wait0->wmma lockstep.
                v2f bv[8];
                #pragma unroll
                for (int tile = 0; tile < 8; ++tile) {
                    // B (4x16 f32): row K = v + 2*half, col N = lrow;
                    // B[k][t] = x[t+k] => K-pair adjacent in t.
                    const int ti = tb + tile * 16 + lrow + kk + 2 * half;
                    bv[tile].x = xc[ti];
                    bv[tile].y = xc[ti + 1];
                }
                #pragma unroll
                for (int tile = 0; tile < 8; ++tile) {
                    acc[tile] = __builtin_amdgcn_wmma_f32_16x16x4_f32(
                        false, a, false, bv[tile], (short)0, acc[tile],
                        false, false);
                }
            }
        }
        __syncthreads();                    // done reading buf before overwrite
    }

    // Epilogue: C/D layout VGPR j -> M = j + 8*half, N = lrow
    float* orow = out + (size_t)n * F_OUT * OUT_W;
    #pragma unroll
    for (int tile = 0; tile < 8; ++tile) {
        const int t = t0 + tb + tile * 16 + lrow;
        if (t < OUT_W) {
            #pragma unroll
            for (int j = 0; j < 8; ++j) {
                const int f = fr + j + 8 * half;
                orow[(size_t)f * OUT_W + t] = acc[tile][j] + b[f];
            }
        }
    }
}

extern "C" void kernel_launch(void* const* d_in, const int* in_sizes, int n_in,
                              void* d_out, int out_size, void* d_ws, size_t ws_size,
                              hipStream_t stream) {
    const float* x = (const float*)d_in[0];
    const float* w = (const float*)d_in[1];
    const float* b = (const float*)d_in[2];
    float* out = (float*)d_out;
    dim3 grid(W_IN / TILE_T, N_BATCH);   // 16 x 64 blocks
    conv1d_wmma_f32_kernel<<<grid, 256, 0, stream>>>(x, w, b, out);
}